// TGRU_66056597012991
// MI455X (gfx1250) — compile-verified
//
#include <hip/hip_runtime.h>
#include <hip/hip_bf16.h>
#include <math.h>

typedef __attribute__((ext_vector_type(2))) float v2f;
typedef __attribute__((ext_vector_type(8))) float v8f;

#define B_DIM   128
#define T_STEPS 256
#define D_DIM   512
#define U_DIM   1024
#define N3U     3072
#define NBLK2   64          // workgroups in the scan kernel (U_DIM / UC)
#define UC      16          // h-columns owned per workgroup

#define BM 128              // phase-1 block tile M
#define BN 128              // phase-1 block tile N
#define KB 16               // phase-1 K panel depth

// ---------------------------------------------------------------------------
// Workspace init: zero h ping-pong buffers and per-step barrier counters.
// ---------------------------------------------------------------------------
__global__ void tgru_init_ws(float* __restrict__ h0, float* __restrict__ h1,
                             unsigned* __restrict__ cnt) {
    int i = blockIdx.x * blockDim.x + threadIdx.x;
    if (i < B_DIM * U_DIM) { h0[i] = 0.0f; h1[i] = 0.0f; }
    if (i < T_STEPS) cnt[i] = 0u;
}

// ---------------------------------------------------------------------------
// Phase 1: x_proj = sequences @ kernel + input_bias
// GEMM M=32768, N=3072, K=512 via V_WMMA_F32_16X16X4_F32.
// 128x128 block tile, double-buffered LDS. 8 waves; wave w owns M band
// [16w,16w+16) and all eight 16-col N tiles (8 accumulators). LDS tiles are
// pre-swizzled into the WMMA fragment layout so every fragment is one
// 8-byte-aligned ds_load_b64:
//   A frag (kq, m, lane): pair = Atile[m*16 + (l&15)][4kq + 2*(l>>4) + e]
//   B frag (kq, nt, lane): pair = Btile[4kq + 2*(l>>4) + e][nt*16 + (l&15)]
// ---------------------------------------------------------------------------
__global__ void tgru_xproj_gemm(const float* __restrict__ A,   // [32768, 512]
                                const float* __restrict__ Bm,  // [512, 3072]
                                const float* __restrict__ bias,// [3072]
                                float* __restrict__ C) {       // [32768, 3072]
    __shared__ __align__(16) float sa[2][BM * KB];  // swizzled [kq4][m8][lane32][e2]
    __shared__ __align__(16) float sb[2][KB * BN];  // swizzled [kq4][nt8][lane32][e2]

    const int tid  = threadIdx.x;
    const int w    = tid >> 5;
    const int lane = tid & 31;
    const int row0 = blockIdx.y * BM;
    const int col0 = blockIdx.x * BN;

    v8f acc[8];
    #pragma unroll
    for (int nt = 0; nt < 8; ++nt) acc[nt] = (v8f){};

    auto stage = [&](int kc, int buf) {
        // A tile 128x16: thread -> row r = tid/2, col base (tid&1)*8, 2 float4.
        const int r  = tid >> 1;
        const int c0 = (tid & 1) * 8;
        const int m  = r >> 4;
        const int rl = r & 15;
        #pragma unroll
        for (int q = 0; q < 2; ++q) {
            const int c  = c0 + q * 4;
            const int kq = c >> 2;
            float4 v = *(const float4*)(A + (size_t)(row0 + r) * D_DIM + kc + c);
            float* dst = sa[buf] + ((kq * 8 + m) * 32 + rl) * 2;
            dst[0]  = v.x; dst[1]  = v.y;        // lanes 0-15 pair (K=0,1)
            dst[32] = v.z; dst[33] = v.w;        // lanes 16-31 pair (K=2,3)
        }
        // B tile 16x128: thread -> row rb = tid/16, col base (tid&15)*8, 2 float4.
        const int rb   = tid >> 4;
        const int kqb  = rb >> 2;
        const int e    = rb & 1;
        const int half = (rb >> 1) & 1;
        #pragma unroll
        for (int q = 0; q < 2; ++q) {
            const int c  = (tid & 15) * 8 + q * 4;
            const int nt = c >> 4;
            const int lb = half * 16 + (c & 15);
            float4 u = *(const float4*)(Bm + (size_t)(kc + rb) * N3U + col0 + c);
            float* dstb = sb[buf] + ((kqb * 8 + nt) * 32 + lb) * 2 + e;
            dstb[0] = u.x; dstb[2] = u.y; dstb[4] = u.z; dstb[6] = u.w;
        }
    };

    stage(0, 0);
    int buf = 0;
    for (int kc = 0; kc < D_DIM; kc += KB) {
        __syncthreads();                       // staging of `buf` complete
        if (kc + KB < D_DIM) {
            stage(kc + KB, buf ^ 1);           // overlap with compute below
            if (kc + 2 * KB < D_DIM) {         // warm L2 one more panel ahead
                __builtin_prefetch(A + (size_t)(row0 + (tid >> 1)) * D_DIM + kc + 2 * KB, 0, 1);
                __builtin_prefetch(Bm + (size_t)(kc + 2 * KB + (tid >> 4)) * N3U + col0, 0, 1);
            }
        }
        const v2f* pa = (const v2f*)sa[buf] + w * 32 + lane;   // + kq*256
        const v2f* pb = (const v2f*)sb[buf] + lane;            // + (kq*8+nt)*32
        #pragma unroll
        for (int kq = 0; kq < 4; ++kq) {
            v2f a = pa[kq * 256];
            #pragma unroll
            for (int nt = 0; nt < 8; ++nt) {
                v2f b = pb[(kq * 8 + nt) * 32];
                acc[nt] = __builtin_amdgcn_wmma_f32_16x16x4_f32(
                    false, a, false, b, (short)0, acc[nt], false, false);
            }
        }
        buf ^= 1;
    }

    // Store with input_bias. C layout: M = j + 8*(lane>=16), N = lane%16.
    const int lm   = lane & 15;
    const int n0   = col0 + lm;
    const int mrow = row0 + w * 16 + ((lane >> 4) << 3);
    float bz[8];
    #pragma unroll
    for (int nt = 0; nt < 8; ++nt) bz[nt] = bias[n0 + nt * 16];
    #pragma unroll
    for (int j = 0; j < 8; ++j) {
        float* crow = C + (size_t)(mrow + j) * N3U + n0;
        #pragma unroll
        for (int nt = 0; nt < 8; ++nt)
            crow[nt * 16] = acc[nt][j] + bz[nt];
    }
}

// ---------------------------------------------------------------------------
// Phase 2: persistent cooperative scan. 64 WGs x 256 threads.
// Each WG pins its 1024x48 recurrent-kernel slice in LDS, pre-swizzled into
// the WMMA B-fragment layout: frag(kq, nt, lane) pair at ((kq*3+nt)*32+l)*2.
// Per step: C[128,48] = (h*dec) @ Wslice; gates; ping-pong h; device barrier.
// ---------------------------------------------------------------------------
__global__ void tgru_scan(const float* __restrict__ td,     // [B, T]
                          const float* __restrict__ rk,     // [U, 3U]
                          const float* __restrict__ rbias,  // [3U]
                          const float* __restrict__ xproj,  // [B, T, 3U]
                          float* __restrict__ h0,
                          float* __restrict__ h1,
                          unsigned* __restrict__ cnt,       // [T]
                          float* __restrict__ out) {        // [B, U]
    extern __shared__ float smem[];
    float* wlds = smem;                         // [1024*48] swizzled
    float* dec  = smem + U_DIM * 48;            // [128]

    const int tid  = threadIdx.x;
    const int w    = tid >> 5;
    const int lane = tid & 31;
    const int lm   = lane & 15;
    const int koff = (lane >> 4) * 2;
    const int u0   = blockIdx.x * UC;

    // Stage + swizzle the W slice once. Output index i decodes as:
    // e = i&1, lane = (i>>1)&31, nt = (i>>6)%3, kq = i/192
    // source row = 4*kq + 2*(lane>>4) + e ; source col = nt*1024 + u0 + (lane&15)
    for (int i = tid; i < U_DIM * 48; i += 256) {
        const int e  = i & 1;
        const int l  = (i >> 1) & 31;
        const int nt = (i >> 6) % 3;
        const int kq = i / 192;
        const int r  = 4 * kq + ((l >> 4) << 1) + e;
        wlds[i] = rk[(size_t)r * N3U + nt * U_DIM + u0 + (l & 15)];
    }
    const float rb_z = rbias[u0 + lm];
    const float rb_r = rbias[U_DIM + u0 + lm];
    const float rb_h = rbias[2 * U_DIM + u0 + lm];
    __syncthreads();

    const int   arow  = w * 16 + lm;            // A-fragment row (batch index)
    const int   u     = u0 + lm;                // output column
    const int   mbase = w * 16 + ((lane >> 4) << 3);
    const v2f*  wfrag = (const v2f*)wlds + lane; // + (kq*3+nt)*32
    float* hbuf[2] = { h0, h1 };

    for (int t = 0; t < T_STEPS; ++t) {
        const float* hin  = hbuf[t & 1];
        float*       hout = hbuf[(t + 1) & 1];

        if (tid < B_DIM)
            dec[tid] = 1.0f / logf(2.71828182845904523f + td[(size_t)tid * T_STEPS + t]);
        __syncthreads();

        v8f az = {}, ar = {}, ah = {};
        const float  dscale = dec[arow];
        const float* hrow   = hin + (size_t)arow * U_DIM;

        for (int kq = 0; kq < U_DIM / 4; ++kq) {
            float2 hv = *(const float2*)(hrow + 4 * kq + koff);
            v2f a;
            a.x = hv.x * dscale;
            a.y = hv.y * dscale;
            const v2f* wq = wfrag + kq * 96;     // (kq*3+nt)*32
            v2f b0 = wq[0];
            v2f b1 = wq[32];
            v2f b2 = wq[64];
            az = __builtin_amdgcn_wmma_f32_16x16x4_f32(false, a, false, b0,
                                                       (short)0, az, false, false);
            ar = __builtin_amdgcn_wmma_f32_16x16x4_f32(false, a, false, b1,
                                                       (short)0, ar, false, false);
            ah = __builtin_amdgcn_wmma_f32_16x16x4_f32(false, a, false, b2,
                                                       (short)0, ah, false, false);
        }

        // Gates + state update for this wave's rows / this WG's columns.
        #pragma unroll
        for (int j = 0; j < 8; ++j) {
            const int    b_  = mbase + j;
            const size_t xo  = ((size_t)b_ * T_STEPS + t) * N3U;
            const float  xz  = xproj[xo + u];
            const float  xr  = xproj[xo + U_DIM + u];
            const float  xh  = xproj[xo + 2 * U_DIM + u];
            const float  hdv = hin[(size_t)b_ * U_DIM + u] * dec[b_];
            const float  z   = 1.0f / (1.0f + __expf(-(xz + az[j] + rb_z)));
            const float  r   = 1.0f / (1.0f + __expf(-(xr + ar[j] + rb_r)));
            const float  hh  = tanhf(xh + r * (ah[j] + rb_h));
            const float  hn  = z * hdv + (1.0f - z) * hh;
            hout[(size_t)b_ * U_DIM + u] = hn;
            if (t == T_STEPS - 1) out[(size_t)b_ * U_DIM + u] = hn;
        }

        // Device-scope arrive-and-spin barrier (per-step counter, no reset).
        __syncthreads();
        if (tid == 0) {
            __threadfence();
            atomicAdd(&cnt[t], 1u);
            while (__hip_atomic_load(&cnt[t], __ATOMIC_RELAXED,
                                     __HIP_MEMORY_SCOPE_AGENT) < (unsigned)NBLK2) {
                __builtin_amdgcn_s_sleep(1);
            }
            __threadfence();
        }
        __syncthreads();
    }
}

// ---------------------------------------------------------------------------
extern "C" void kernel_launch(void* const* d_in, const int* in_sizes, int n_in,
                              void* d_out, int out_size, void* d_ws, size_t ws_size,
                              hipStream_t stream) {
    const float* seq  = (const float*)d_in[0];  // [B, T, D]
    const float* td   = (const float*)d_in[1];  // [B, T]
    const float* kern = (const float*)d_in[2];  // [D, 3U]
    const float* rk   = (const float*)d_in[3];  // [U, 3U]
    const float* ib   = (const float*)d_in[4];  // [3U]
    const float* rb   = (const float*)d_in[5];  // [3U]
    float* out = (float*)d_out;

    char* ws = (char*)d_ws;
    float*    h0    = (float*)(ws);                       // 512 KB
    float*    h1    = (float*)(ws + 524288);              // 512 KB
    unsigned* cnt   = (unsigned*)(ws + 1048576);          // 1 KB
    float*    xproj = (float*)(ws + 1052672);             // 402.7 MB

    tgru_init_ws<<<512, 256, 0, stream>>>(h0, h1, cnt);

    dim3 g1(N3U / BN, (B_DIM * T_STEPS) / BM);            // 24 x 256
    tgru_xproj_gemm<<<g1, 256, 0, stream>>>(seq, kern, ib, xproj);

    size_t smem = (size_t)(U_DIM * 48 + B_DIM) * sizeof(float);  // ~192.5 KB
    tgru_scan<<<NBLK2, 256, smem, stream>>>(td, rk, rb, xproj, h0, h1, cnt, out);
}